// SparseModel_72610717106765
// MI455X (gfx1250) — compile-verified
//
#include <hip/hip_runtime.h>

// CDNA5 / gfx1250 — wave32, WMMA (no MFMA).
typedef __attribute__((ext_vector_type(2))) float v2f;
typedef __attribute__((ext_vector_type(8))) float v8f;

constexpr int kBatch = 4096;
constexpr int kNNZ   = 32;
constexpr int kDim   = 2048;
constexpr int kTilesPerWave  = 4;                       // 4 x 16 = 64 columns per wave
constexpr int kColsPerWave   = 16 * kTilesPerWave;
constexpr int kColsPerBlock  = 8 * kColsPerWave;        // 8 waves -> 512 columns

// out[b, :] = sum_k values[b,k] * kernel[indices[b,k], :] + bias
//
// Densified-tile WMMA formulation: per (16 batch rows x 16 cols) tile,
// C = A(16x512) * G(512x16), A block-diagonal (A[m, 32m+k] = values[m,k]),
// K consumed 4 at a time with V_WMMA_F32_16X16X4_F32 (fp32 accumulate).
// Each wave carries 4 adjacent column tiles so the A fragment, the k-row
// selects, and the two gathered-row base addresses amortize over 4 WMMAs;
// tiles 1..3 load via immediate offsets (+64B/+128B/+192B).
//
// Fragment layouts per CDNA5 ISA 7.12.2:
//   A 16x4 f32 : VGPR0 = {K0 lanes0-15, K2 lanes16-31}, VGPR1 = {K1, K3}, M = lane&15
//   B 4x16 f32 : VGPR0 = {K0 row lanes0-15, K2 row lanes16-31}, VGPR1 = {K1, K3}, N = lane&15
//   C 16x16 f32: VGPR j = {M=j lanes0-15, M=8+j lanes16-31}, N = lane&15
__global__ __launch_bounds__(256) void sparse_gather_wmma_kernel(
    const float* __restrict__ values,   // [B, NNZ]
    const int*   __restrict__ indices,  // [B, NNZ]
    const float* __restrict__ kern,     // [D, D]
    const float* __restrict__ bias,     // [D]
    float*       __restrict__ out)      // [B, D]
{
    const int lane = threadIdx.x & 31;
    const int wave = threadIdx.x >> 5;
    const int ml   = lane & 15;   // M (for A) / N (for B,C) position
    const int half = lane >> 4;   // 0: K0/K1 chunk half, 1: K2/K3 half

    const int colBase = blockIdx.x * kColsPerBlock + wave * kColsPerWave;
    const unsigned nc = (unsigned)(colBase + ml);       // lane's column in tile 0
    const int rowBase = blockIdx.y * 16;
    const int baseEl  = rowBase * kNNZ;                 // tile's first (row, nnz) element

    v8f acc0 = {}, acc1 = {}, acc2 = {}, acc3 = {};

    // ---- Pipeline stage for t = 0 (uniform chunk -> SMEM; B rows -> VMEM) ----
    int4   i4 = *reinterpret_cast<const int4*>(indices + baseEl);
    float4 v4 = *reinterpret_cast<const float4*>(values  + baseEl);

    const float* r0 = kern + (((unsigned)(half ? i4.z : i4.x)) << 11) + nc;
    const float* r1 = kern + (((unsigned)(half ? i4.w : i4.y)) << 11) + nc;
    v2f b0 = {r0[0],  r1[0]};
    v2f b1 = {r0[16], r1[16]};
    v2f b2 = {r0[32], r1[32]};
    v2f b3 = {r0[48], r1[48]};

    // t = m*8 + c encodes (batch row m in tile, 4-wide nnz chunk c); offset = t*4.
    for (int t = 0; t < 128; ++t) {
        // A fragment for current t: only batch row m = t>>3 nonzero in this K chunk.
        const int m = t >> 3;
        v2f a;
        a.x = (ml == m) ? (half ? v4.z : v4.x) : 0.0f;
        a.y = (ml == m) ? (half ? v4.w : v4.y) : 0.0f;

        // ---- Stage t+1: scalars + gathered-row loads (overlap with WMMAs) ----
        int4   ni4 = i4;
        float4 nv4 = v4;
        v2f nb0 = b0, nb1 = b1, nb2 = b2, nb3 = b3;
        if (t + 1 < 128) {
            const int off = baseEl + (t + 1) * 4;
            ni4 = *reinterpret_cast<const int4*>(indices + off);
            nv4 = *reinterpret_cast<const float4*>(values  + off);
            const float* p0 = kern + (((unsigned)(half ? ni4.z : ni4.x)) << 11) + nc;
            const float* p1 = kern + (((unsigned)(half ? ni4.w : ni4.y)) << 11) + nc;
            nb0 = v2f{p0[0],  p1[0]};
            nb1 = v2f{p0[16], p1[16]};
            nb2 = v2f{p0[32], p1[32]};
            nb3 = v2f{p0[48], p1[48]};
        }

        // ---- 4 WMMAs sharing one A fragment (256 useful MACs) ----
        // (neg_a, A, neg_b, B, c_mod, C, reuse_a, reuse_b)
        acc0 = __builtin_amdgcn_wmma_f32_16x16x4_f32(false, a, false, b0, (short)0, acc0, false, false);
        acc1 = __builtin_amdgcn_wmma_f32_16x16x4_f32(false, a, false, b1, (short)0, acc1, false, false);
        acc2 = __builtin_amdgcn_wmma_f32_16x16x4_f32(false, a, false, b2, (short)0, acc2, false, false);
        acc3 = __builtin_amdgcn_wmma_f32_16x16x4_f32(false, a, false, b3, (short)0, acc3, false, false);

        b0 = nb0; b1 = nb1; b2 = nb2; b3 = nb3;
        i4 = ni4; v4 = nv4;
    }

    // ---- Epilogue: fused bias add, fp32 stores ----
    const float bv0 = bias[nc];
    const float bv1 = bias[nc + 16];
    const float bv2 = bias[nc + 32];
    const float bv3 = bias[nc + 48];
    const int outRow = rowBase + half * 8;
#pragma unroll
    for (int j = 0; j < 8; ++j) {
        float* orow = out + (size_t)(outRow + j) * kDim + nc;
        orow[0]  = acc0[j] + bv0;
        orow[16] = acc1[j] + bv1;
        orow[32] = acc2[j] + bv2;
        orow[48] = acc3[j] + bv3;
    }
}

extern "C" void kernel_launch(void* const* d_in, const int* in_sizes, int n_in,
                              void* d_out, int out_size, void* d_ws, size_t ws_size,
                              hipStream_t stream) {
    const float* values  = (const float*)d_in[0];
    const int*   indices = (const int*)  d_in[1];
    const float* kern    = (const float*)d_in[2];
    const float* bias    = (const float*)d_in[3];
    float* outp = (float*)d_out;

    dim3 grid(kDim / kColsPerBlock, kBatch / 16);   // (4, 256)
    dim3 block(256);
    hipLaunchKernelGGL(sparse_gather_wmma_kernel, grid, block, 0, stream,
                       values, indices, kern, bias, outp);
}